// PointnetSAModuleMSG_WithSampling_41248865910987
// MI455X (gfx1250) — compile-verified
//
#include <hip/hip_runtime.h>
#include <hip/hip_bf16.h>
#include <stdint.h>

#define BB 8
#define NN 16384
#define CC 64
#define MM 1024

typedef float v2f __attribute__((ext_vector_type(2)));
typedef float v8f __attribute__((ext_vector_type(8)));

__constant__ constexpr float BN_INV_C = 0.99999500003749973f;

__device__ __forceinline__ v8f wmma_f32(v2f a, v2f b, v8f c) {
  // D(16x16) = A(16x4) * B(4x16) + C   (fp32 WMMA, 8-arg form)
  return __builtin_amdgcn_wmma_f32_16x16x4_f32(false, a, false, b, (short)0, c,
                                               false, false);
}

// ---------------------------------------------------------------- FPS ------
__global__ __launch_bounds__(1024) void fps_kernel(
    const float* __restrict__ xyz, float* __restrict__ newxyz) {
  __shared__ float rv[32];
  __shared__ int   ri[32];
  __shared__ float bxs, bys, bzs;
  __shared__ int   sel[MM];
  const int b = blockIdx.x, tid = threadIdx.x;
  const int wid = tid >> 5, lane = tid & 31;
  const float* px = xyz + (size_t)b * NN * 3;

  float X[16], Y[16], Z[16], md[16];
#pragma unroll
  for (int j = 0; j < 16; ++j) {
    int p = tid + j * 1024;
    X[j] = px[p * 3 + 0];
    Y[j] = px[p * 3 + 1];
    Z[j] = px[p * 3 + 2];
    md[j] = 1e10f;
  }
  if (tid == 0) { sel[0] = 0; bxs = px[0]; bys = px[1]; bzs = px[2]; }
  __syncthreads();
  float lx = bxs, ly = bys, lz = bzs;

  for (int s = 1; s < MM; ++s) {
    float bv = -1.f; int bi = tid;
#pragma unroll
    for (int j = 0; j < 16; ++j) {
      float dx = X[j] - lx, dy = Y[j] - ly, dz = Z[j] - lz;
      float d = dx * dx + dy * dy + dz * dz;
      md[j] = fminf(md[j], d);
      if (md[j] > bv) { bv = md[j]; bi = tid + j * 1024; }
    }
#pragma unroll
    for (int o = 16; o > 0; o >>= 1) {
      float ov = __shfl_xor(bv, o);
      int   oi = __shfl_xor(bi, o);
      if (ov > bv || (ov == bv && oi < bi)) { bv = ov; bi = oi; }
    }
    if (lane == 0) { rv[wid] = bv; ri[wid] = bi; }
    __syncthreads();
    if (wid == 0) {
      bv = rv[lane]; bi = ri[lane];
#pragma unroll
      for (int o = 16; o > 0; o >>= 1) {
        float ov = __shfl_xor(bv, o);
        int   oi = __shfl_xor(bi, o);
        if (ov > bv || (ov == bv && oi < bi)) { bv = ov; bi = oi; }
      }
      if (lane == 0) {
        sel[s] = bi;
        bxs = px[bi * 3 + 0]; bys = px[bi * 3 + 1]; bzs = px[bi * 3 + 2];
      }
    }
    __syncthreads();
    lx = bxs; ly = bys; lz = bzs;
  }
  const int gi = sel[tid];
  float* o = newxyz + (size_t)(b * MM + tid) * 3;
  o[0] = px[gi * 3 + 0];
  o[1] = px[gi * 3 + 1];
  o[2] = px[gi * 3 + 2];
}

// ---------------------------------------------------------- ball query -----
__global__ __launch_bounds__(128) void ballquery_kernel(
    const float* __restrict__ xyz, const float* __restrict__ newxyz,
    int* __restrict__ gq0, int* __restrict__ gq1) {
  const int wid = threadIdx.x >> 5, lane = threadIdx.x & 31;
  const int cid = blockIdx.x * 4 + wid;
  const int b = cid >> 10;
  const float* nx = newxyz + (size_t)cid * 3;
  const float cx = nx[0], cy = nx[1], cz = nx[2];
  const float* px = xyz + (size_t)b * NN * 3;
  int* o0 = gq0 + (size_t)cid * 16;
  int* o1 = gq1 + (size_t)cid * 32;

  int cnt0 = 0, cnt1 = 0, first0 = -1, first1 = -1;
  const unsigned below = (1u << lane) - 1u;
  for (int base = 0; base < NN; base += 32) {
    const int p = base + lane;
    float dx = px[p * 3 + 0] - cx;
    float dy = px[p * 3 + 1] - cy;
    float dz = px[p * 3 + 2] - cz;
    float d2 = dx * dx + dy * dy + dz * dz;
    bool h0 = d2 < 0.25f, h1 = d2 < 1.0f;
    unsigned m0 = (unsigned)__ballot(h0);
    unsigned m1 = (unsigned)__ballot(h1);
    if (cnt0 < 16) {
      if (first0 < 0 && m0) first0 = base + __ffs(m0) - 1;
      int pos = cnt0 + __popc(m0 & below);
      if (h0 && pos < 16) o0[pos] = p;
      cnt0 += __popc(m0);
    }
    if (cnt1 < 32) {
      if (first1 < 0 && m1) first1 = base + __ffs(m1) - 1;
      int pos = cnt1 + __popc(m1 & below);
      if (h1 && pos < 32) o1[pos] = p;
      cnt1 += __popc(m1);
    }
    if (cnt0 >= 16 && cnt1 >= 32) break;
  }
  const int pad0 = (first0 < 0) ? (NN - 1) : first0;
  const int pad1 = (first1 < 0) ? (NN - 1) : first1;
  if (lane < 16 && lane >= cnt0) o0[lane] = pad0;
  if (lane >= cnt1) o1[lane] = pad1;
}

// -------------------------------------------------- WMMA GEMM helpers ------
template <int CIN, int CINREAL>
__device__ __forceinline__ v8f gemm_tile_lds(const float* __restrict__ W,
                                             const float* __restrict__ bufin,
                                             int trow, int lane) {
  v8f acc = {};
  const int col = lane & 15;
  const int koff = (lane >> 4) << 1;
  const int row = trow + col;
#pragma unroll
  for (int k = 0; k < CIN; k += 4) {
    const int kk = k + koff;
    v2f a, bf;
    a.x = (kk     < CINREAL) ? W[row * CINREAL + kk]     : 0.f;
    a.y = (kk + 1 < CINREAL) ? W[row * CINREAL + kk + 1] : 0.f;
    bf.x = bufin[kk * 17 + col];
    bf.y = bufin[(kk + 1) * 17 + col];
    acc = wmma_f32(a, bf, acc);
  }
  return acc;
}

__device__ __forceinline__ void act_store(v8f acc, const float* __restrict__ g,
                                          const float* __restrict__ bb, int trow,
                                          float* __restrict__ bufout, int lane) {
  const int col = lane & 15;
  const int roff = trow + ((lane >> 4) << 3);
#pragma unroll
  for (int r = 0; r < 8; ++r) {
    const int row = roff + r;
    float y = acc[r] * BN_INV_C * g[row] + bb[row];
    bufout[row * 17 + col] = fmaxf(y, 0.f);
  }
}

// ------------------------------------------- grouped MLP + max-pool --------
// One wave per center; 16 sample-columns per pass; fp32 WMMA through 3 layers.
template <int NS, int CMID>
__global__ __launch_bounds__(128) void mlp_msg_kernel(
    const float* __restrict__ xyz, const float* __restrict__ feat,
    const float* __restrict__ newxyz, const int* __restrict__ gidx,
    const float* __restrict__ w0, const float* __restrict__ g0, const float* __restrict__ b0,
    const float* __restrict__ w1, const float* __restrict__ g1, const float* __restrict__ b1,
    const float* __restrict__ w2, const float* __restrict__ g2, const float* __restrict__ b2,
    float* __restrict__ out_nf, int ch_off) {
  __shared__ float bufA[4][96 * 17];  // input (68 rows) then L1 output (<=96)
  __shared__ float bufB[4][64 * 17];  // L0 output
  const int wid = threadIdx.x >> 5, lane = threadIdx.x & 31;
  const int cid = blockIdx.x * 4 + wid;
  const int b = cid >> 10, m = cid & 1023;
  float* bA = bufA[wid];
  float* bBf = bufB[wid];
  const float* nx = newxyz + (size_t)cid * 3;
  const float cx = nx[0], cy = nx[1], cz = nx[2];
  const int col = lane & 15, half = lane >> 4;

  float fmx[64];
#pragma unroll
  for (int i = 0; i < 64; ++i) fmx[i] = 0.f;  // post-ReLU values are >= 0

  for (int p = 0; p < NS / 16; ++p) {
    // ---- build input tile: rows 0-2 rel xyz, 3-66 features, 67 zero pad
    const int gi = gidx[(size_t)cid * NS + p * 16 + col];
    const float* pz = xyz + (size_t)(b * NN + gi) * 3;
    if (half == 0) {
      bA[0 * 17 + col] = pz[0] - cx;
      bA[1 * 17 + col] = pz[1] - cy;
      bA[2 * 17 + col] = pz[2] - cz;
    } else {
      bA[67 * 17 + col] = 0.f;
    }
    const float* fb = feat + (size_t)b * CC * NN + gi;
#pragma unroll
    for (int c = 0; c < 32; ++c) {
      const int ch = half * 32 + c;
      bA[(3 + ch) * 17 + col] = fb[(size_t)ch * NN];
    }
    // ---- L0: 67 -> 64
    for (int t = 0; t < 4; ++t) {
      v8f acc = gemm_tile_lds<68, 67>(w0, bA, t * 16, lane);
      act_store(acc, g0, b0, t * 16, bBf, lane);
    }
    // ---- L1: 64 -> CMID
    for (int t = 0; t < CMID / 16; ++t) {
      v8f acc = gemm_tile_lds<64, 64>(w1, bBf, t * 16, lane);
      act_store(acc, g1, b1, t * 16, bA, lane);
    }
    // ---- L2: CMID -> 128, fold running max (over samples) in registers
    for (int t = 0; t < 8; ++t) {
      v8f acc = gemm_tile_lds<CMID, CMID>(w2, bA, t * 16, lane);
      const int roff = t * 16 + (half << 3);
#pragma unroll
      for (int r = 0; r < 8; ++r) {
        const int row = roff + r;
        float y = fmaxf(acc[r] * BN_INV_C * g2[row] + b2[row], 0.f);
        fmx[t * 8 + r] = fmaxf(fmx[t * 8 + r], y);
      }
    }
  }
  // max over the 16 columns of each half-wave
#pragma unroll
  for (int t = 0; t < 8; ++t) {
#pragma unroll
    for (int r = 0; r < 8; ++r) {
      float v = fmx[t * 8 + r];
      v = fmaxf(v, __shfl_xor(v, 1));
      v = fmaxf(v, __shfl_xor(v, 2));
      v = fmaxf(v, __shfl_xor(v, 4));
      v = fmaxf(v, __shfl_xor(v, 8));
      fmx[t * 8 + r] = v;
    }
  }
  if (col == 0) {  // lanes 0 and 16 write low/high row halves
#pragma unroll
    for (int t = 0; t < 8; ++t)
#pragma unroll
      for (int r = 0; r < 8; ++r) {
        const int row = t * 16 + (half << 3) + r;
        out_nf[(size_t)b * 256 * MM + (size_t)(ch_off + row) * MM + m] =
            fmx[t * 8 + r];
      }
  }
}

// ------------------------------------------------ confidence head ----------
__global__ __launch_bounds__(128) void conf_kernel(
    const float* __restrict__ nf, const float* __restrict__ cw0,
    const float* __restrict__ cg0, const float* __restrict__ cb0,
    const float* __restrict__ cw1, const float* __restrict__ cb1,
    float* __restrict__ out_conf) {
  const int wid = threadIdx.x >> 5, lane = threadIdx.x & 31;
  const int col = lane & 15, half = lane >> 4, koff = half * 2;
  const int j = (blockIdx.x * 4 + wid) * 16 + col;  // 0..8191
  const int b = j >> 10, m = j & 1023;
  const float* nfc = nf + (size_t)b * 256 * MM + m;

  float part = 0.f;
  for (int t = 0; t < 8; ++t) {
    v8f acc = {};
    const int row = t * 16 + col;
#pragma unroll 4
    for (int k = 0; k < 256; k += 4) {
      const int kk = k + koff;
      v2f a, bf;
      a.x = cw0[row * 256 + kk];
      a.y = cw0[row * 256 + kk + 1];
      bf.x = nfc[(size_t)kk * MM];
      bf.y = nfc[(size_t)(kk + 1) * MM];
      acc = wmma_f32(a, bf, acc);
    }
    const int roff = t * 16 + (half << 3);
#pragma unroll
    for (int r = 0; r < 8; ++r) {
      const int rw = roff + r;
      float h = fmaxf(acc[r] * BN_INV_C * cg0[rw] + cb0[rw], 0.f);
      part += h * cw1[rw];
    }
  }
  part += __shfl_xor(part, 16);  // other half holds the other 8 rows / tile
  if (half == 0) out_conf[j] = part + cb1[0];
}

// --------------------------------------------------------------- launch ----
extern "C" void kernel_launch(void* const* d_in, const int* in_sizes, int n_in,
                              void* d_out, int out_size, void* d_ws,
                              size_t ws_size, hipStream_t stream) {
  (void)in_sizes; (void)n_in; (void)out_size; (void)ws_size;
  const float* xyz  = (const float*)d_in[0];
  const float* feat = (const float*)d_in[1];
  const float *w00 = (const float*)d_in[2],  *g00 = (const float*)d_in[3],  *b00 = (const float*)d_in[4];
  const float *w01 = (const float*)d_in[5],  *g01 = (const float*)d_in[6],  *b01 = (const float*)d_in[7];
  const float *w02 = (const float*)d_in[8],  *g02 = (const float*)d_in[9],  *b02 = (const float*)d_in[10];
  const float *w10 = (const float*)d_in[11], *g10 = (const float*)d_in[12], *b10 = (const float*)d_in[13];
  const float *w11 = (const float*)d_in[14], *g11 = (const float*)d_in[15], *b11 = (const float*)d_in[16];
  const float *w12 = (const float*)d_in[17], *g12 = (const float*)d_in[18], *b12 = (const float*)d_in[19];
  const float *cw0 = (const float*)d_in[20], *cg0 = (const float*)d_in[21], *cb0 = (const float*)d_in[22];
  const float *cw1 = (const float*)d_in[23], *cb1 = (const float*)d_in[24];

  float* out        = (float*)d_out;
  float* out_newxyz = out;                        // (8,1024,3)   24576
  float* out_nf     = out + BB * MM * 3;          // (8,256,1024) 2097152
  float* out_conf   = out_nf + BB * 256 * MM;     // (8,1024,1)   8192

  int* gq0 = (int*)d_ws;                          // (8,1024,16)
  int* gq1 = gq0 + BB * MM * 16;                  // (8,1024,32)

  fps_kernel<<<BB, 1024, 0, stream>>>(xyz, out_newxyz);
  ballquery_kernel<<<BB * MM / 4, 128, 0, stream>>>(xyz, out_newxyz, gq0, gq1);
  mlp_msg_kernel<16, 64><<<BB * MM / 4, 128, 0, stream>>>(
      xyz, feat, out_newxyz, gq0, w00, g00, b00, w01, g01, b01, w02, g02, b02,
      out_nf, 0);
  mlp_msg_kernel<32, 96><<<BB * MM / 4, 128, 0, stream>>>(
      xyz, feat, out_newxyz, gq1, w10, g10, b10, w11, g11, b11, w12, g12, b12,
      out_nf, 128);
  // 8192 columns / (4 waves * 16 cols per wave) = 128 blocks
  conf_kernel<<<(BB * MM) / 64, 128, 0, stream>>>(
      out_nf, cw0, cg0, cb0, cw1, cb1, out_conf);
}